// I2Pool_326417514934
// MI455X (gfx1250) — compile-verified
//
#include <hip/hip_runtime.h>

// Problem constants (match the reference): x[B,C,H,W], guide[B,1,H,W], f32.
namespace {
constexpr int B = 32, C = 256, H = 96, W = 96;
constexpr int WAVES = 8;                 // 256 threads / wave32
constexpr int CH_PER_WAVE = C / WAVES;   // 32 channels per wave
}

// CDNA5 async global->LDS path (ASYNCcnt). Guarded so compile never breaks.
#if __has_builtin(__builtin_amdgcn_global_load_async_to_lds_b32) && \
    __has_builtin(__builtin_amdgcn_s_wait_asynccnt)
#define USE_ASYNC_LDS 1
#else
#define USE_ASYNC_LDS 0
#endif

// ---- wave32 max-scan primitives (ds_bpermute lane-permute HW, no LDS mem) --
__device__ __forceinline__ float lane_up_max(float v, int d, int lane) {
  int src = lane - d;
  float t = __int_as_float(
      __builtin_amdgcn_ds_bpermute(src << 2, __float_as_int(v)));
  return fmaxf(v, src < 0 ? v : t);  // identity (self) when out of range
}

__device__ __forceinline__ float inclusive_max_scan32(float v, int lane) {
  v = lane_up_max(v, 1, lane);
  v = lane_up_max(v, 2, lane);
  v = lane_up_max(v, 4, lane);
  v = lane_up_max(v, 8, lane);
  v = lane_up_max(v, 16, lane);
  return v;
}

__device__ __forceinline__ float bcast31(float v) {
  return __int_as_float(__builtin_amdgcn_readlane(__float_as_int(v), 31));
}

// Scan one 96-element row held as (lane, lane+32, lane+64) and store it.
__device__ __forceinline__ void scan_and_store(float v0, float v1, float v2,
                                               int lane,
                                               float* __restrict__ orow) {
  float s0 = inclusive_max_scan32(v0, lane);
  float carry = bcast31(s0);                          // max of elems [0,32)
  float s1 = fmaxf(inclusive_max_scan32(v1, lane), carry);
  carry = bcast31(s1);                                // max of elems [0,64)
  float s2 = fmaxf(inclusive_max_scan32(v2, lane), carry);
  orow[lane]      = s0;
  orow[lane + 32] = s1;
  orow[lane + 64] = s2;
}

#if USE_ASYNC_LDS
typedef __attribute__((address_space(1))) int as1_int;
typedef __attribute__((address_space(3))) int as3_int;

// Issue 3 coalesced async b32 copies: one 384B row, global -> LDS.
__device__ __forceinline__ void async_row_to_lds(const float* g, float* l,
                                                 int lane) {
  __builtin_amdgcn_global_load_async_to_lds_b32(
      (as1_int*)(g + lane), (as3_int*)(l + lane), 0, 0);
  __builtin_amdgcn_global_load_async_to_lds_b32(
      (as1_int*)(g + lane + 32), (as3_int*)(l + lane + 32), 0, 0);
  __builtin_amdgcn_global_load_async_to_lds_b32(
      (as1_int*)(g + lane + 64), (as3_int*)(l + lane + 64), 0, 0);
}
#endif

__global__ __launch_bounds__(256) void I2Pool_326417514934_kernel(
    const float* __restrict__ x, const float* __restrict__ guide,
    float* __restrict__ out) {
  const int lane = threadIdx.x & 31;
  const int wave = threadIdx.x >> 5;
  const int bh = blockIdx.x;           // one (b,h) per block
  const int b = bh / H;
  const int h = bh - b * H;

  // Guide row: loaded once into registers, reused for all 256 channels.
  const float* grow = guide + ((size_t)b * H + h) * (size_t)W;
  const float g0 = grow[lane];
  const float g1 = grow[lane + 32];
  const float g2 = grow[lane + 64];

  const size_t chan_stride = (size_t)H * W;                    // 9216
  const size_t base = (size_t)b * C * chan_stride + (size_t)h * W;
  const int c0 = wave * CH_PER_WAVE;

#if USE_ASYNC_LDS
  // Per-wave double buffer: 2 x 96 floats; 8 waves -> 6 KB of 320 KB LDS.
  __shared__ float lds_buf[WAVES * 2 * W];
  float* lbuf = &lds_buf[wave * 2 * W];

  async_row_to_lds(x + base + (size_t)c0 * chan_stride, lbuf, lane);
  int p = 0;
  for (int j = 0; j < CH_PER_WAVE; ++j) {
    if (j + 1 < CH_PER_WAVE) {
      // Prefetch next channel's row into the other buffer, then wait for the
      // current buffer's 3 loads (in-order completion => oldest 3 done).
      async_row_to_lds(x + base + (size_t)(c0 + j + 1) * chan_stride,
                       lbuf + (p ^ 1) * W, lane);
      __builtin_amdgcn_s_wait_asynccnt(3);
    } else {
      __builtin_amdgcn_s_wait_asynccnt(0);
    }
    const float* cur = lbuf + p * W;
    float v0 = cur[lane] * g0;
    float v1 = cur[lane + 32] * g1;
    float v2 = cur[lane + 64] * g2;
    float* orow = out + base + (size_t)(c0 + j) * chan_stride;
    scan_and_store(v0, v1, v2, lane, orow);
    p ^= 1;
  }
#else
  for (int j = 0; j < CH_PER_WAVE; ++j) {
    const float* row = x + base + (size_t)(c0 + j) * chan_stride;
    if (j + 1 < CH_PER_WAVE) {
      // global_prefetch_b8: lanes span the next row's 3 cachelines.
      __builtin_prefetch(row + chan_stride + (size_t)lane * 3, 0, 0);
    }
    float v0 = row[lane] * g0;
    float v1 = row[lane + 32] * g1;
    float v2 = row[lane + 64] * g2;
    float* orow = out + base + (size_t)(c0 + j) * chan_stride;
    scan_and_store(v0, v1, v2, lane, orow);
  }
#endif
}

extern "C" void kernel_launch(void* const* d_in, const int* in_sizes, int n_in,
                              void* d_out, int out_size, void* d_ws,
                              size_t ws_size, hipStream_t stream) {
  (void)in_sizes; (void)n_in; (void)out_size; (void)d_ws; (void)ws_size;
  const float* x = (const float*)d_in[0];      // [B,C,H,W] f32
  const float* guide = (const float*)d_in[1];  // [B,1,H,W] f32
  float* out = (float*)d_out;                  // [B,C,H,W] f32
  dim3 grid(B * H);                            // 3072 blocks
  dim3 block(WAVES * 32);                      // 256 threads = 8 wave32
  I2Pool_326417514934_kernel<<<grid, block, 0, stream>>>(x, guide, out);
}